// QuadriLinearScore_1580547965137
// MI455X (gfx1250) — compile-verified
//
#include <hip/hip_runtime.h>
#include <hip/hip_bf16.h>

// Problem constants (from the reference)
#define RANK    396
#define WEMB    768
#define TAGSET  50
#define TAGEMB  20
#define BATCH   8
#define SEQ     512
#define NROWS   (BATCH * (SEQ - 1))     // 4088 flattened (n, a) positions
#define NCOLS   (TAGSET * TAGSET)       // 2500 flattened (b, c) tag pairs
#define INV_SQRT_RANK 0.050251890762960605f  // 1/sqrt(396)

typedef __attribute__((ext_vector_type(2))) float v2f;
typedef __attribute__((ext_vector_type(4))) float v4f;
typedef __attribute__((ext_vector_type(8))) float v8f;

// Native fp32 WMMA: D(16x16,f32) = A(16x4,f32) * B(4x16,f32) + C
__device__ __forceinline__ v8f wmma_f32(v2f a, v2f b, v8f c) {
    return __builtin_amdgcn_wmma_f32_16x16x4_f32(
        /*neg_a=*/false, a, /*neg_b=*/false, b,
        /*c_mod=*/(short)0, c, /*reuse_a=*/false, /*reuse_b=*/false);
}

__device__ __forceinline__ int iclamp(int v, int hi) { return v > hi ? hi : v; }

// ---------------------------------------------------------------------------
// Kernel 1: g2 = tag_emd @ V, g3 = tag_emd @ W   (50x20 x 20x396 — tiny)
// ---------------------------------------------------------------------------
__global__ void g23_kernel(const float* __restrict__ tag,
                           const float* __restrict__ V,
                           const float* __restrict__ W,
                           float* __restrict__ g2,
                           float* __restrict__ g3) {
    int idx = blockIdx.x * blockDim.x + threadIdx.x;
    if (idx >= TAGSET * RANK) return;
    int b = idx / RANK;
    int k = idx - b * RANK;
    float s2 = 0.f, s3 = 0.f;
#pragma unroll
    for (int t = 0; t < TAGEMB; ++t) {
        float e = tag[b * TAGEMB + t];
        s2 += e * V[t * RANK + k];
        s3 += e * W[t * RANK + k];
    }
    g2[idx] = s2;
    g3[idx] = s3;
}

// ---------------------------------------------------------------------------
// Kernel 2: P[b*50+c, k] = g2[b,k] * g3[c,k] / sqrt(RANK)   ([2500, 396])
// Row-major over k == B^T panels for the stage-3 GEMM (contiguous B-frags).
// ---------------------------------------------------------------------------
__global__ void pmat_kernel(const float* __restrict__ g2,
                            const float* __restrict__ g3,
                            float* __restrict__ P) {
    int idx = blockIdx.x * blockDim.x + threadIdx.x;
    if (idx >= NCOLS * RANK) return;
    int bc = idx / RANK;
    int k  = idx - bc * RANK;
    int b  = bc / TAGSET;
    int c  = bc - b * TAGSET;
    P[idx] = g2[b * RANK + k] * g3[c * RANK + k] * INV_SQRT_RANK;
}

// ---------------------------------------------------------------------------
// Kernel 3: transpose [WEMB][RANK] -> [RANK][WEMB] so temp01's B-frags are
// contiguous float2 loads instead of stride-396 gathers.
// ---------------------------------------------------------------------------
__global__ void transpose_kernel(const float* __restrict__ src,
                                 float* __restrict__ dst) {
    int idx = blockIdx.x * blockDim.x + threadIdx.x;
    if (idx >= WEMB * RANK) return;
    int n = idx / WEMB;           // dst row  (rank index)
    int k = idx - n * WEMB;       // dst col  (wemb index)
    dst[idx] = src[(size_t)k * RANK + n];   // coalesced writes
}

// ---------------------------------------------------------------------------
// Kernel 4: temp01[r, :] = (we[n,a,:] @ U) * (we[n,a+1,:] @ T), r = n*511 + a
// 2x2 register-blocked: each wave owns a 32x32 output block; two WMMA chains
// (U and T) share the K = 768 loop -> 8 WMMAs per K-step sharing 8 frag loads.
// Grid maps exactly: 128 row-blocks x 13 col-blocks = 208 blocks x 8 waves.
// ---------------------------------------------------------------------------
__global__ void __launch_bounds__(256) temp01_kernel(
        const float* __restrict__ we,
        const float* __restrict__ Ut,   // [RANK][WEMB]
        const float* __restrict__ Tt,   // [RANK][WEMB]
        float* __restrict__ t01) {
    const int lane = threadIdx.x & 31;
    const int wave = threadIdx.x >> 5;
    const int tile = blockIdx.x * 8 + wave;    // 0 .. 1663, exact
    const int rowB = tile / 13;
    const int colB = tile - rowB * 13;
    const int row0 = rowB * 32;
    const int col0 = colB * 32;

    const int half = lane >> 4;   // 0: K,K+1   1: K+2,K+3
    const int m    = lane & 15;

    // A row pointers per M sub-tile (clamped; edges masked at store)
    const float* x0p[2];
    const float* x1p[2];
#pragma unroll
    for (int mt = 0; mt < 2; ++mt) {
        int r  = iclamp(row0 + mt * 16 + m, NROWS - 1);
        int nb = r / (SEQ - 1);
        int a  = r - nb * (SEQ - 1);
        x0p[mt] = we + ((size_t)nb * SEQ + a) * WEMB;   // word_emb[n, a,   :]
        x1p[mt] = x0p[mt] + WEMB;                        // word_emb[n, a+1, :]
    }
    // B column pointers per N sub-tile (transposed weights: contiguous pairs)
    const float* bup[2];
    const float* btp[2];
#pragma unroll
    for (int nt = 0; nt < 2; ++nt) {
        int n  = iclamp(col0 + nt * 16 + m, RANK - 1);
        bup[nt] = Ut + (size_t)n * WEMB;
        btp[nt] = Tt + (size_t)n * WEMB;
    }

    v8f c0[2][2] = {};   // X0 @ U
    v8f c1[2][2] = {};   // X1 @ T
    for (int kk = 0; kk < WEMB; kk += 4) {
        const int ka = kk + 2 * half;
        v2f a0[2], a1[2], bu[2], bt[2];
#pragma unroll
        for (int mt = 0; mt < 2; ++mt) {
            a0[mt] = *(const v2f*)(x0p[mt] + ka);
            a1[mt] = *(const v2f*)(x1p[mt] + ka);
        }
#pragma unroll
        for (int nt = 0; nt < 2; ++nt) {
            bu[nt] = *(const v2f*)(bup[nt] + ka);
            bt[nt] = *(const v2f*)(btp[nt] + ka);
        }
#pragma unroll
        for (int mt = 0; mt < 2; ++mt)
#pragma unroll
            for (int nt = 0; nt < 2; ++nt) {
                c0[mt][nt] = wmma_f32(a0[mt], bu[nt], c0[mt][nt]);
                c1[mt][nt] = wmma_f32(a1[mt], bt[nt], c1[mt][nt]);
            }
    }

#pragma unroll
    for (int mt = 0; mt < 2; ++mt)
#pragma unroll
        for (int nt = 0; nt < 2; ++nt)
#pragma unroll
            for (int i = 0; i < 8; ++i) {
                int gr = row0 + mt * 16 + i + 8 * half;
                int gc = col0 + nt * 16 + m;
                if (gr < NROWS && gc < RANK)
                    t01[(size_t)gr * RANK + gc] = c0[mt][nt][i] * c1[mt][nt][i];
            }
}

// ---------------------------------------------------------------------------
// Kernel 5: score[r, bc] = sum_k temp01[r,k] * P[bc,k]
// GEMM [4088,396] x [396,2500]. Each block stages one 64-row A band (64x396
// floats = 101 KB) into LDS cooperatively; its 8 waves each compute a 64x64
// output block of the same band (4x4 WMMA tiles, 16 accumulators), so A-frags
// come from ds_load_b64 and B-frags are contiguous float2 rows of P.
// Grid maps exactly: 64 row bands x 5 col groups; 8 waves = 40 col blocks.
// ---------------------------------------------------------------------------
#define A_BAND_FLOATS (64 * RANK)     // 25344 floats = 101376 bytes of LDS

__global__ void __launch_bounds__(256) score_kernel(
        const float* __restrict__ t01,
        const float* __restrict__ P,
        float* __restrict__ out) {
    extern __shared__ float lds[];     // A_BAND_FLOATS, passed at launch

    const int rowBand  = blockIdx.x / 5;           // 0..63
    const int colGroup = blockIdx.x - rowBand * 5; // 0..4
    const int R0 = rowBand * 64;

    // ---- cooperative stage: A band [64][396] -> LDS, float4 granularity ----
    {
        const int nVec4 = A_BAND_FLOATS / 4;       // 6336 float4s, 99 per row
        for (int e = threadIdx.x; e < nVec4; e += 256) {
            int r  = e / 99;                       // row within band
            int kq = e - r * 99;                   // float4 index within row
            int gr = iclamp(R0 + r, NROWS - 1);    // clamp: edge rows masked later
            v4f v = *(const v4f*)(t01 + (size_t)gr * RANK + 4 * kq);
            *(v4f*)(lds + r * RANK + 4 * kq) = v;
        }
    }
    __syncthreads();

    // ---- compute: one 64x64 output block per wave ----
    const int lane = threadIdx.x & 31;
    const int wave = threadIdx.x >> 5;
    const int C0 = (colGroup * 8 + wave) * 64;     // 0..2496

    const int half = lane >> 4;
    const int m    = lane & 15;

    const float* brow[4];
#pragma unroll
    for (int j = 0; j < 4; ++j) {
        int bc = iclamp(C0 + j * 16 + m, NCOLS - 1);
        brow[j] = P + (size_t)bc * RANK;           // row of P == column of B
    }
    const float* arow[4];
#pragma unroll
    for (int i = 0; i < 4; ++i)
        arow[i] = lds + (i * 16 + m) * RANK;

    v8f acc[4][4] = {};
    for (int kk = 0; kk < RANK; kk += 4) {
        const int ka = kk + 2 * half;
        v2f af[4], bf[4];
#pragma unroll
        for (int i = 0; i < 4; ++i)
            af[i] = *(const v2f*)(arow[i] + ka);   // ds_load_b64 from LDS
#pragma unroll
        for (int j = 0; j < 4; ++j)
            bf[j] = *(const v2f*)(brow[j] + ka);   // contiguous global float2
#pragma unroll
        for (int i = 0; i < 4; ++i)
#pragma unroll
            for (int j = 0; j < 4; ++j)
                acc[i][j] = wmma_f32(af[i], bf[j], acc[i][j]);
    }

#pragma unroll
    for (int i = 0; i < 4; ++i)
#pragma unroll
        for (int j = 0; j < 4; ++j)
#pragma unroll
            for (int ii = 0; ii < 8; ++ii) {
                int gr = R0 + i * 16 + ii + 8 * half;
                int gc = C0 + j * 16 + m;
                if (gr < NROWS && gc < NCOLS)
                    out[(size_t)gr * NCOLS + gc] = acc[i][j][ii];
            }
}

// ---------------------------------------------------------------------------
extern "C" void kernel_launch(void* const* d_in, const int* in_sizes, int n_in,
                              void* d_out, int out_size, void* d_ws, size_t ws_size,
                              hipStream_t stream) {
    const float* we  = (const float*)d_in[0];   // [8, 512, 768]
    const float* tag = (const float*)d_in[1];   // [50, 20]
    const float* T   = (const float*)d_in[2];   // [768, 396]
    const float* U   = (const float*)d_in[3];   // [768, 396]
    const float* V   = (const float*)d_in[4];   // [20, 396]
    const float* W   = (const float*)d_in[5];   // [20, 396]
    float* out = (float*)d_out;                 // [8, 511, 50, 50]

    // Workspace (floats): g2 | g3 | P | temp01 | Ut | Tt  => ~13.0 MB
    float* g2  = (float*)d_ws;
    float* g3  = g2  + (size_t)TAGSET * RANK;        // +19800
    float* P   = g3  + (size_t)TAGSET * RANK;        // +19800
    float* t01 = P   + (size_t)NCOLS * RANK;         // +990000 (t01 16B-aligned)
    float* Ut  = t01 + (size_t)NROWS * RANK;         // +1618848
    float* Tt  = Ut  + (size_t)WEMB * RANK;          // +304128

    {
        int n = TAGSET * RANK;                       // 19800
        g23_kernel<<<(n + 255) / 256, 256, 0, stream>>>(tag, V, W, g2, g3);
    }
    {
        int n = NCOLS * RANK;                        // 990000
        pmat_kernel<<<(n + 255) / 256, 256, 0, stream>>>(g2, g3, P);
    }
    {
        int n = WEMB * RANK;                         // 304128 = 1188*256
        transpose_kernel<<<(n + 255) / 256, 256, 0, stream>>>(U, Ut);
        transpose_kernel<<<(n + 255) / 256, 256, 0, stream>>>(T, Tt);
    }
    {
        // 128 row-blocks(32) x 13 col-blocks(32) = 1664 waves = 208 blocks
        temp01_kernel<<<208, 256, 0, stream>>>(we, Ut, Tt, t01);
    }
    {
        // 64 row bands(64) x 5 col groups(8 waves x 64 cols) = 320 blocks
        score_kernel<<<320, 256, A_BAND_FLOATS * sizeof(float), stream>>>(t01, P, out);
    }
}